// ChanelDevParcelLoss_13477607375448
// MI455X (gfx1250) — compile-verified
//
#include <hip/hip_runtime.h>
#include <hip/hip_bf16.h>
#include <math.h>

// Problem constants (from reference setup_inputs)
#define N_IMG    8
#define C_CH     80
#define HW       65536          // 256*256
#define NCLS     20
#define CNUM     4
#define NPARCEL  8192
#define IGNORE_INDEX 255

// Workspace layout (float indices)
#define OFF_COLMAX 0                               // [640]  per-(n,c) spatial max
#define OFF_COLSUM 640                             // [640]  per-(n,c) sum of exp(x-max)
#define OFF_PART   1280                            // [0]=nll_sum [1]=valid_cnt [2]=div_sum
#define OFF_SEGSUM 2048                            // [8192*20] parcel class sums
#define OFF_SEGCNT (OFF_SEGSUM + NPARCEL*NCLS)     // [8192] parcel pixel counts (float)
#define OFF_TGT    (OFF_SEGCNT + NPARCEL)          // [8192] parcel target (int, init -1)
#define WS_TOTAL   (OFF_TGT + NPARCEL)             // 182272 floats (~712 KB)

// ---------------- CDNA5 async global->LDS path (guarded) ----------------
#if defined(__AMDGCN__) && __has_builtin(__builtin_amdgcn_global_load_async_to_lds_b128)
  #define HAVE_ASYNC 1
#else
  #define HAVE_ASYNC 0
#endif

#if HAVE_ASYNC
typedef __attribute__((ext_vector_type(4))) int v4i_t;
typedef __attribute__((address_space(1))) v4i_t gas_v4i;   // global (AS1)
typedef __attribute__((address_space(3))) v4i_t las_v4i;   // LDS (AS3)

__device__ __forceinline__ void async_copy16(const float* g, float* l) {
  // each lane copies 16B from the global address g to the LDS address l, tracked by ASYNCcnt
  __builtin_amdgcn_global_load_async_to_lds_b128(
      (gas_v4i*)g, (las_v4i*)l, /*imm offset*/0, /*cpol*/0);
}
#if __has_builtin(__builtin_amdgcn_s_wait_asynccnt)
  #define ASYNC_WAIT(n) __builtin_amdgcn_s_wait_asynccnt(n)
#else
  #define ASYNC_WAIT(n) asm volatile("s_wait_asynccnt %0" :: "n"(n) : "memory")
#endif
#endif

// ---------------- block reduction (wave32) ----------------
__device__ __forceinline__ float block_sum256(float v) {
  __shared__ float sm[8];
  #pragma unroll
  for (int o = 16; o > 0; o >>= 1) v += __shfl_down(v, o, 32);
  const int lane = threadIdx.x & 31, w = threadIdx.x >> 5;
  if (lane == 0) sm[w] = v;
  __syncthreads();
  v = (threadIdx.x < 8) ? sm[threadIdx.x] : 0.0f;
  if (w == 0) {
    #pragma unroll
    for (int o = 4; o > 0; o >>= 1) v += __shfl_down(v, o, 32);
  }
  __syncthreads();           // allow sm reuse by a later call
  return v;                  // valid in thread 0
}

// ---------------- kernels (k_col_lse first so disasm snippet shows async ops) ----

// One block per (n,c) column: online logsumexp stats over HW pixels.
// Streams the 256KB column through LDS with double-buffered async copies.
__global__ void __launch_bounds__(256) k_col_lse(const float* __restrict__ F, float* ws) {
  const int col = blockIdx.x;             // n*C_CH + c
  const int t   = threadIdx.x;
  const float* src = F + (size_t)col * HW;
  float m = -__builtin_inff(), s = 0.0f;

#if HAVE_ASYNC
  __shared__ __align__(16) float buf[2 * 1024];   // 2 stages x 1024 floats (4KB)
  async_copy16(src + t * 4, &buf[t * 4]);         // prime stage 0
  for (int stg = 0; stg < 64; ++stg) {
    const int cur = stg & 1;
    if (stg < 63) {
      async_copy16(src + (size_t)(stg + 1) * 1024 + t * 4,
                   &buf[((stg + 1) & 1) * 1024 + t * 4]);
      ASYNC_WAIT(1);                              // stage `stg` landed; next may be in flight
    } else {
      ASYNC_WAIT(0);
    }
    // each lane reads back exactly the 16B it copied -> no barrier required
    const float4 x4 = *reinterpret_cast<const float4*>(&buf[cur * 1024 + t * 4]);
    const float xs[4] = {x4.x, x4.y, x4.z, x4.w};
    #pragma unroll
    for (int j = 0; j < 4; ++j) {
      const float x = xs[j];
      if (x <= m) { s += expf(x - m); }
      else        { s = s * expf(m - x) + 1.0f; m = x; }
    }
  }
#else
  const float4* s4 = reinterpret_cast<const float4*>(src);
  for (int stg = 0; stg < 64; ++stg) {
    const float4 x4 = s4[(size_t)stg * 256 + t];
    const float xs[4] = {x4.x, x4.y, x4.z, x4.w};
    #pragma unroll
    for (int j = 0; j < 4; ++j) {
      const float x = xs[j];
      if (x <= m) { s += expf(x - m); }
      else        { s = s * expf(m - x) + 1.0f; m = x; }
    }
  }
#endif

  // combine (m,s) across the wave32
  #pragma unroll
  for (int o = 16; o > 0; o >>= 1) {
    const float m2 = __shfl_down(m, o, 32);
    const float s2 = __shfl_down(s, o, 32);
    const float M  = fmaxf(m, m2);
    s = s * expf(m - M) + s2 * expf(m2 - M);
    m = M;
  }
  __shared__ float wm[8], wss[8];
  const int lane = t & 31, w = t >> 5;
  if (lane == 0) { wm[w] = m; wss[w] = s; }
  __syncthreads();
  if (t == 0) {
    float M = wm[0], S = wss[0];
    #pragma unroll
    for (int i = 1; i < 8; ++i) {
      const float M2 = fmaxf(M, wm[i]);
      S = S * expf(M - M2) + wss[i] * expf(wm[i] - M2);
      M = M2;
    }
    ws[OFF_COLMAX + col] = M;
    ws[OFF_COLSUM + col] = S;
  }
}

__global__ void k_init(float* ws) {
  const int i = blockIdx.x * blockDim.x + threadIdx.x;
  if (i >= WS_TOTAL) return;
  if (i < OFF_COLSUM)        ws[i] = -__builtin_inff();  // colmax
  else if (i < OFF_SEGSUM)   ws[i] = 0.0f;               // colsum + partials + pad
  else if (i < OFF_TGT)      ws[i] = 0.0f;               // seg_sum + seg_cnt
  else ((int*)ws)[i] = -1;                               // parcel target (segment_max init)
}

// Four pixels per thread: per-channel b128 loads, channel-group max-pool ->
// parcel scatter + diversity sum.
__global__ void __launch_bounds__(256) k_main(const float* __restrict__ F,
                                              const int* __restrict__ target,
                                              const int* __restrict__ parcel,
                                              float* ws) {
  float* part     = ws + OFF_PART;
  float* seg_sum  = ws + OFF_SEGSUM;
  float* seg_cnt  = ws + OFF_SEGCNT;
  int*   tgtp     = (int*)(ws + OFF_TGT);

  const int tid4 = blockIdx.x * 256 + threadIdx.x;   // handles pixels [4*tid4, 4*tid4+3]
  const int pix0 = tid4 * 4;                         // global pixel over n*hw
  const int n    = pix0 >> 16;                       // HW = 65536; const within block
  const int t    = threadIdx.x;

  __shared__ float cm[C_CH], csi[C_CH];
  if (t < C_CH) {
    cm[t]  = ws[OFF_COLMAX + n * C_CH + t];
    csi[t] = 1.0f / ws[OFF_COLSUM + n * C_CH + t];
  }
  __syncthreads();

  const size_t base = (size_t)n * C_CH * HW + (pix0 & (HW - 1));
  const int4 tg4 = reinterpret_cast<const int4*>(target)[tid4];
  const int4 p4  = reinterpret_cast<const int4*>(parcel)[tid4];
  const int tg[4] = {tg4.x, tg4.y, tg4.z, tg4.w};
  const int pp[4] = {p4.x,  p4.y,  p4.z,  p4.w};
  bool valid[4];
  #pragma unroll
  for (int i = 0; i < 4; ++i) valid[i] = (tg[i] != IGNORE_INDEX);

  float divacc = 0.0f;
  for (int k = 0; k < NCLS; ++k) {
    float dmax[4], vmax[4];
    #pragma unroll
    for (int i = 0; i < 4; ++i) { dmax[i] = -__builtin_inff(); vmax[i] = -__builtin_inff(); }
    #pragma unroll
    for (int j = 0; j < CNUM; ++j) {
      const int c = k * CNUM + j;
      const float4 x4 = *reinterpret_cast<const float4*>(&F[base + (size_t)c * HW]);
      const float xs[4] = {x4.x, x4.y, x4.z, x4.w};
      const float cmc = cm[c], ci = csi[c];
      #pragma unroll
      for (int i = 0; i < 4; ++i) {
        dmax[i] = fmaxf(dmax[i], xs[i]);
        vmax[i] = fmaxf(vmax[i], expf(xs[i] - cmc) * ci);
      }
    }
    #pragma unroll
    for (int i = 0; i < 4; ++i) {
      if (valid[i]) atomicAdd(&seg_sum[pp[i] * NCLS + k], dmax[i]);
      divacc += vmax[i];
    }
  }
  #pragma unroll
  for (int i = 0; i < 4; ++i) {
    if (valid[i]) {
      atomicAdd(&seg_cnt[pp[i]], 1.0f);
      atomicMax(&tgtp[pp[i]], tg[i]);
    }
  }
  const float bsum = block_sum256(divacc);
  if (t == 0) atomicAdd(&part[2], bsum);
}

// One thread per parcel: mean -> log_softmax -> NLL, reduced into partials.
__global__ void __launch_bounds__(256) k_parcel(float* ws) {
  float* part          = ws + OFF_PART;
  const float* seg_sum = ws + OFF_SEGSUM;
  const float* seg_cnt = ws + OFF_SEGCNT;
  const int*   tgtp    = (const int*)(ws + OFF_TGT);

  const int p = blockIdx.x * 256 + threadIdx.x;    // 32 blocks x 256 = 8192
  const float cnt = seg_cnt[p];
  const float inv = 1.0f / fmaxf(cnt, 1.0f);

  float mean[NCLS];
  float mx = -__builtin_inff();
  #pragma unroll
  for (int k = 0; k < NCLS; ++k) {
    mean[k] = seg_sum[p * NCLS + k] * inv;
    mx = fmaxf(mx, mean[k]);
  }
  float se = 0.0f;
  #pragma unroll
  for (int k = 0; k < NCLS; ++k) se += expf(mean[k] - mx);
  const float lse = logf(se) + mx;

  int tg = tgtp[p];
  tg = tg < 0 ? 0 : (tg > NCLS - 1 ? NCLS - 1 : tg);
  float tv = 0.0f;                                  // predicated select: no scratch
  #pragma unroll
  for (int k = 0; k < NCLS; ++k) tv = (k == tg) ? mean[k] : tv;

  const float vld = (cnt > 0.0f) ? 1.0f : 0.0f;
  const float nll = (lse - tv) * vld;

  const float nllSum = block_sum256(nll);
  const float vldSum = block_sum256(vld);
  if (threadIdx.x == 0) {
    atomicAdd(&part[0], nllSum);
    atomicAdd(&part[1], vldSum);
  }
}

__global__ void k_final(const float* ws, float* out) {
  if (threadIdx.x == 0 && blockIdx.x == 0) {
    const float* part = ws + OFF_PART;
    out[0] = part[0] / fmaxf(part[1], 1.0f);                         // loss_dis
    out[1] = 1.0f - (part[2] / (float)(N_IMG * NCLS)) / (float)NCLS; // loss_div
  }
}

extern "C" void kernel_launch(void* const* d_in, const int* in_sizes, int n_in,
                              void* d_out, int out_size, void* d_ws, size_t ws_size,
                              hipStream_t stream) {
  const float* F      = (const float*)d_in[0];
  const int*   target = (const int*)d_in[1];
  const int*   parcel = (const int*)d_in[2];
  float* ws  = (float*)d_ws;
  float* out = (float*)d_out;

  k_init   <<<(WS_TOTAL + 255) / 256,  256, 0, stream>>>(ws);
  k_col_lse<<<N_IMG * C_CH,            256, 0, stream>>>(F, ws);        // 640 blocks
  k_main   <<<(N_IMG * HW) / (256*4),  256, 0, stream>>>(F, target, parcel, ws); // 512 blocks
  k_parcel <<<NPARCEL / 256,           256, 0, stream>>>(ws);           // 32 blocks
  k_final  <<<1, 32, 0, stream>>>(ws, out);
}